// RS_eval_32796370272792
// MI455X (gfx1250) — compile-verified
//
#include <hip/hip_runtime.h>
#include <stdint.h>

// ---- MI455X / gfx1250: wave32, WMMA 16x16x4 f32, TDM tensor load with LDS padding ----

typedef float v8f  __attribute__((ext_vector_type(8)));
typedef float v2f  __attribute__((ext_vector_type(2)));
typedef unsigned int v4u __attribute__((ext_vector_type(4)));
typedef int   v8i  __attribute__((ext_vector_type(8)));
typedef int   v4i  __attribute__((ext_vector_type(4)));

#define Bn   8
#define Cn   64
#define Hn   128
#define Wn   128
#define NUMK 32
#define NTHREADS 1024     // 32 wave32s per workgroup
#define PADW 132          // lds_x row stride: 2 zero guard cols each side
#define XOFF 2
#define RROWS 132         // lds_r rows: 2 zero guard rows each side
#define NXF  (Hn * PADW)  // lds_x floats
#define NRF  (RROWS * Wn) // lds_r floats

__global__ __launch_bounds__(NTHREADS) void nms_topk_kernel(
    const float* __restrict__ x, float* __restrict__ out)
{
    extern __shared__ unsigned char smem[];
    float* lds_x = (float*)smem;                 // padded X image, later padded WS
    float* lds_r = lds_x + NXF;                  // row-pass R (guard rows), later keys
    unsigned long long* lds_red =
        (unsigned long long*)(smem + (NXF + NRF) * sizeof(float));
    int* lds_top = (int*)(smem + (NXF + NRF) * sizeof(float)
                               + 32 * sizeof(unsigned long long));

    const int tid  = threadIdx.x;
    const int lane = tid & 31;
    const int wv   = tid >> 5;
    const int bc   = blockIdx.x;                 // one workgroup per (b,c) image
    const float* img = x + (size_t)bc * (Hn * Wn);

    // ---------------- Stage -1: zero halos (whole lds_x + lds_r guard rows) -------------
    for (int p = tid; p < NXF; p += NTHREADS) lds_x[p] = 0.0f;
    for (int p = tid; p < 2 * Wn; p += NTHREADS) {
        lds_r[p] = 0.0f;                         // rows -2,-1
        lds_r[(2 + Hn) * Wn + p] = 0.0f;         // rows 128,129
    }
    __syncthreads();

    // ---------------- Stage 0: TDM async load into padded LDS layout --------------------
    // pad_enable: after every 128 DWORDs insert 4 DWORDs -> LDS row stride 528 B = PADW*4.
    // lds_addr points at interior col 2, so guard cols (pre-zeroed) frame each row.
    if (wv == 0) {
        uint64_t ga = (uint64_t)(uintptr_t)img;
        uint32_t lds_off = (uint32_t)(uintptr_t)lds_x + XOFF * 4u;
        v4u g0;
        g0.x = 1u;                                       // count=1, user mode, no gather
        g0.y = lds_off;                                  // lds_addr (interior)
        g0.z = (uint32_t)ga;                             // global_addr[31:0]
        g0.w = (uint32_t)((ga >> 32) & 0x1FFFFFFu) | (2u << 30); // addr[56:32] | type=2
        v8i g1;
        g1[0] = (int)((2u << 16)      // data_size = 4B
                    | (1u << 20)      // pad_enable
                    | (6u << 22)      // pad_interval code 6 = 128 DWORDs
                    | (3u << 25));    // pad_amount  code 3 = 4 DWORDs
        g1[1] = (int)(128u << 16);    // tensor_dim0 lo16 = 128
        g1[2] = (int)(128u << 16);    // tensor_dim0 hi=0 | tensor_dim1 lo16 = 128
        g1[3] = (int)(128u << 16);    // tensor_dim1 hi=0 | tile_dim0 = 128
        g1[4] = (int)(128u);          // tile_dim1 = 128, tile_dim2 = 0
        g1[5] = (int)(128u);          // tensor_dim0_stride = 128 elements
        g1[6] = 0;
        g1[7] = 0;
        v4i g2 = {0, 0, 0, 0};
        v4i g3 = {0, 0, 0, 0};
        v8i g4 = {0, 0, 0, 0, 0, 0, 0, 0};
        __builtin_amdgcn_tensor_load_to_lds(g0, g1, g2, g3, g4, 0);
        __builtin_amdgcn_s_wait_tensorcnt(0);
    }
    __syncthreads();

    const int m16 = lane & 15;
    const int hi  = lane >> 4;

    // ---------------- Stage 1: horizontal box-sum as WMMA banded matmul  R = X * H ------
    // Unconditional halo loads: s in [-2, 129] maps to padded cols [0, 131].
    for (int t = wv; t < 64; t += 32) {
        const int ti = t >> 3, tj = t & 7;
        const int row = ti * 16 + m16;               // A-matrix M index for this lane
        const int n   = tj * 16 + m16;               // B-matrix N index for this lane
        v8f acc = {0, 0, 0, 0, 0, 0, 0, 0};
        #pragma unroll
        for (int m = 0; m < 5; ++m) {
            const int s = tj * 16 - 2 + 4 * m + 2 * hi;  // K pair this lane holds
            v2f a, b;
            a.x = lds_x[row * PADW + XOFF + s];
            a.y = lds_x[row * PADW + XOFF + s + 1];
            const int d0 = s - n, d1 = s + 1 - n;        // tile-invariant band values
            b.x = (d0 >= -1 && d0 <= 1) ? 1.0f : 0.0f;
            b.y = (d1 >= -1 && d1 <= 1) ? 1.0f : 0.0f;
            acc = __builtin_amdgcn_wmma_f32_16x16x4_f32(
                false, a, false, b, (short)0, acc, false, false);
        }
        #pragma unroll
        for (int i = 0; i < 8; ++i)                  // C/D layout: M = i + 8*hi, N = m16
            lds_r[(2 + ti * 16 + i + 8 * hi) * Wn + n] = acc[i];
    }
    __syncthreads();

    // ---------------- Stage 2: vertical box-sum as WMMA banded matmul  WS = V * R -------
    // Unconditional halo loads: k in [-2, 129] maps to guard rows [0, 131].
    for (int t = wv; t < 64; t += 32) {
        const int ti = t >> 3, tj = t & 7;
        const int r = ti * 16 + m16;                 // output row (A's M)
        const int n = tj * 16 + m16;                 // output col (B's N)
        v8f acc = {0, 0, 0, 0, 0, 0, 0, 0};
        #pragma unroll
        for (int m = 0; m < 5; ++m) {
            const int k = ti * 16 - 2 + 4 * m + 2 * hi;
            v2f a, b;
            const int d0 = r - k, d1 = r - (k + 1);      // tile-invariant band values
            a.x = (d0 >= -1 && d0 <= 1) ? 1.0f : 0.0f;
            a.y = (d1 >= -1 && d1 <= 1) ? 1.0f : 0.0f;
            b.x = lds_r[(2 + k) * Wn + n];
            b.y = lds_r[(3 + k) * Wn + n];
            acc = __builtin_amdgcn_wmma_f32_16x16x4_f32(
                false, a, false, b, (short)0, acc, false, false);
        }
        #pragma unroll
        for (int i = 0; i < 8; ++i)                  // WS overwrites X interior in LDS
            lds_x[(ti * 16 + i + 8 * hi) * PADW + XOFF + n] = acc[i];
    }
    __syncthreads();

    // ---------------- Stage 3: peak mask + sortable key ---------------------------------
    // masked = (WS == 3x3 max of WS) ? WS : 0.  Exact bounds checks here: the max-pool
    // pads with -inf (guard zeros must NOT participate), matching the reference.
    {
        uint32_t* S = (uint32_t*)lds_r;
        for (int p = tid; p < Hn * Wn; p += NTHREADS) {
            const int r = p >> 7, c = p & 127;
            const float v = lds_x[r * PADW + XOFF + c];
            float mx = -__builtin_huge_valf();
            #pragma unroll
            for (int dr = -1; dr <= 1; ++dr) {
                #pragma unroll
                for (int dc = -1; dc <= 1; ++dc) {
                    const int rr = r + dr, cc = c + dc;
                    if (rr >= 0 && rr < Hn && cc >= 0 && cc < Wn)
                        mx = fmaxf(mx, lds_x[rr * PADW + XOFF + cc]);
                }
            }
            const float masked = (v == mx) ? v : 0.0f;
            uint32_t ub = __float_as_uint(masked);
            ub = ((int)ub >= 0) ? (ub | 0x80000000u) : ~ub;   // totally ordered uint
            S[p] = ub;
        }
    }
    __syncthreads();

    // ---------------- Stage 4: top-32 via repeated argmax (wave32 shfl reduce) ----------
    // 64-bit key = (sortable << 32) | ~idx  -> ties resolve to the smallest index,
    // matching lax.top_k's stable ordering.
    {
        uint32_t* S = (uint32_t*)lds_r;
        for (int it = 0; it < NUMK; ++it) {
            unsigned long long best = 0ull;
            for (int p = tid; p < Hn * Wn; p += NTHREADS) {
                unsigned long long key =
                    ((unsigned long long)S[p] << 32) | (uint32_t)(~p);
                if (key > best) best = key;
            }
            #pragma unroll
            for (int off = 16; off > 0; off >>= 1) {
                unsigned long long o = __shfl_xor(best, off, 32);
                if (o > best) best = o;
            }
            if (lane == 0) lds_red[wv] = best;
            __syncthreads();
            if (tid == 0) {
                unsigned long long b = 0ull;
                for (int i = 0; i < 32; ++i)
                    if (lds_red[i] > b) b = lds_red[i];
                const int idx = (int)(~(uint32_t)b);
                lds_top[it] = idx;
                S[idx] = 0u;                      // remove winner (0 < any real key)
            }
            __syncthreads();
        }
    }

    // ---------------- Stage 5: gather 3x3 windows (zero-padded) + clipped bboxes --------
    float* out_win = out;                                    // (B,C,NUM,3,3)
    float* out_pos = out + (size_t)Bn * Cn * NUMK * 9;       // (B,C,NUM,4) as floats
    for (int t = tid; t < NUMK * 9; t += NTHREADS) {
        const int nn = t / 9, k = t % 9;
        const int idx = lds_top[nn];
        const int h = idx >> 7, w = idx & 127;
        const int rr = h + k / 3 - 1, cc = w + (k % 3) - 1;
        const float v = (rr >= 0 && rr < Hn && cc >= 0 && cc < Wn)
                            ? img[rr * Wn + cc] : 0.0f;
        out_win[(size_t)bc * (NUMK * 9) + t] = v;
    }
    for (int t = tid; t < NUMK * 4; t += NTHREADS) {
        const int nn = t >> 2, comp = t & 3;
        const int idx = lds_top[nn];
        const int h = idx >> 7, w = idx & 127;
        int v;
        switch (comp) {
            case 0:  v = w - 1; break;   // x1
            case 1:  v = h - 1; break;   // y1
            case 2:  v = w + 1; break;   // x2
            default: v = h + 1; break;   // y2
        }
        v = v < 0 ? 0 : (v > Wn - 1 ? Wn - 1 : v);
        out_pos[(size_t)bc * (NUMK * 4) + t] = (float)v;
    }
}

extern "C" void kernel_launch(void* const* d_in, const int* in_sizes, int n_in,
                              void* d_out, int out_size, void* d_ws, size_t ws_size,
                              hipStream_t stream) {
    (void)in_sizes; (void)n_in; (void)out_size; (void)d_ws; (void)ws_size;
    const float* x = (const float*)d_in[0];   // (8,64,128,128) f32
    float* out = (float*)d_out;               // 147456 win + 65536 pos
    constexpr size_t SMEM = (size_t)(NXF + NRF) * sizeof(float)  // padded X/WS + R/keys
                          + 32 * sizeof(unsigned long long)      // cross-wave reduce
                          + 32 * sizeof(int);                    // top indices
    nms_topk_kernel<<<dim3(Bn * Cn), dim3(NTHREADS), SMEM, stream>>>(x, out);
}